// GatingNetworkMoE_24000277250500
// MI455X (gfx1250) — compile-verified
//
#include <hip/hip_runtime.h>

typedef float v2f __attribute__((ext_vector_type(2)));
typedef float v4f __attribute__((ext_vector_type(4)));
typedef float v8f __attribute__((ext_vector_type(8)));

constexpr int TOKENS = 16384;
constexpr int DIM    = 2048;
constexpr int NE     = 16;     // experts
constexpr int WAVES  = 8;      // waves per block
constexpr int TILE   = 16;     // tokens per wave (one 16x16 WMMA C tile)

__global__ __launch_bounds__(WAVES * 32)
void moe_gate_kernel(const float* __restrict__ x,
                     const float* __restrict__ noise,
                     const float* __restrict__ W,
                     const float* __restrict__ bias,
                     float* __restrict__ out)
{
    __shared__ float s_logits[WAVES][TILE][NE + 1];  // +1 pad: kill bank conflicts

    const int lane = threadIdx.x & 31;
    const int wave = threadIdx.x >> 5;
    const int tile = blockIdx.x * WAVES + wave;      // 16-token tile index
    const int row0 = tile * TILE;

    // f32 WMMA A(16x4)/B(4x16) layout: lanes 0-15 carry K=0,1 ; lanes 16-31 carry K=2,3
    const int m  = lane & 15;   // token-in-tile for A, expert for B, expert(N) for C/D
    const int hi = lane >> 4;   // half-wave select
    const int kh = hi * 2;

    const float* xr = x + (size_t)(row0 + m) * DIM + kh;   // A stream (the 134 MB one)
    const float* wr = W + (size_t)m * DIM + kh;            // B stream (128 KB, cache-resident)

    v8f acc = {};
    for (int k = 0; k < DIM; k += 16) {
        v2f a0 = *(const v2f*)(xr + k +  0);
        v2f b0 = *(const v2f*)(wr + k +  0);
        v2f a1 = *(const v2f*)(xr + k +  4);
        v2f b1 = *(const v2f*)(wr + k +  4);
        v2f a2 = *(const v2f*)(xr + k +  8);
        v2f b2 = *(const v2f*)(wr + k +  8);
        v2f a3 = *(const v2f*)(xr + k + 12);
        v2f b3 = *(const v2f*)(wr + k + 12);
        __builtin_prefetch(xr + k + 128, 0, 1);   // global_prefetch_b8 on the big stream
        acc = __builtin_amdgcn_wmma_f32_16x16x4_f32(false, a0, false, b0, (short)0, acc, false, false);
        acc = __builtin_amdgcn_wmma_f32_16x16x4_f32(false, a1, false, b1, (short)0, acc, false, false);
        acc = __builtin_amdgcn_wmma_f32_16x16x4_f32(false, a2, false, b2, (short)0, acc, false, false);
        acc = __builtin_amdgcn_wmma_f32_16x16x4_f32(false, a3, false, b3, (short)0, acc, false, false);
    }

    // C/D layout: reg r, lanes 0-15 -> token r, expert=lane ; lanes 16-31 -> token r+8, expert=lane-16
    const float be = bias[m];
    const float* np = noise + (size_t)(row0 + hi * 8) * NE + m;
    #pragma unroll
    for (int r = 0; r < 8; ++r) {
        float v = acc[r] + be + 0.1f * np[(size_t)r * NE];
        s_logits[wave][r + hi * 8][m] = v;
    }
    __syncthreads();

    if (lane < TILE) {
        // top-2 with lowest-index tie-break (matches jax.lax.top_k)
        float best = -INFINITY, second = -INFINITY;
        int bi = 0, si = 0;
        #pragma unroll
        for (int e = 0; e < NE; ++e) {
            float v = s_logits[wave][lane][e];
            if (v > best)        { second = best; si = bi; best = v; bi = e; }
            else if (v > second) { second = v; si = e; }
        }
        // softmax over (best, second): numerically identical to exp(v-max)/sum
        float e2  = __expf(second - best);
        float inv = 1.0f / (1.0f + e2);
        float w1  = inv;
        float w2  = e2 * inv;

        float* orow = out + (size_t)(row0 + lane) * NE;
        v4f z = {};
        *(v4f*)(orow +  0) = z;
        *(v4f*)(orow +  4) = z;
        *(v4f*)(orow +  8) = z;
        *(v4f*)(orow + 12) = z;
        orow[bi] = w1;          // same-wave stores to same addr are ordered (ISA 7.3)
        orow[si] = w2;
    }
}

extern "C" void kernel_launch(void* const* d_in, const int* in_sizes, int n_in,
                              void* d_out, int out_size, void* d_ws, size_t ws_size,
                              hipStream_t stream) {
    const float* x     = (const float*)d_in[0];
    const float* noise = (const float*)d_in[1];
    const float* W     = (const float*)d_in[2];
    const float* b     = (const float*)d_in[3];
    float* out         = (float*)d_out;

    dim3 block(WAVES * 32);                       // 256 threads = 8 wave32
    dim3 grid(TOKENS / (WAVES * TILE));           // 16384 / 128 = 128 blocks
    moe_gate_kernel<<<grid, block, 0, stream>>>(x, noise, W, b, out);
}